// PolyOrderModule_48756468744532
// MI455X (gfx1250) — compile-verified
//
#include <hip/hip_runtime.h>

#define GRID_G 14
#define PATCH_P 16
#define BATCH 32
#define CH 96
#define HW 224  // GRID_G * PATCH_P

typedef __attribute__((ext_vector_type(2))) float v2f;
typedef __attribute__((ext_vector_type(4))) float v4f;
typedef __attribute__((ext_vector_type(8))) float v8f;

// ---------------------------------------------------------------------------
// Pass 1: one workgroup per (b, pi).  sq[b*256 + pi*16 + pj] =
//         sum over (c, gi, gj) of x[b, c, gi*16+pi, gj*16+pj]^2.
// XDL reduction: D = ones(16x4) @ B(4x16) + C -> every D row is the running
// per-pj sum in exact f32.  Addressing is fully incremental: advancing one
// "unit" stream by 32 units is off += 7232 elements, plus +3360 when gj
// wraps (the gi 14->0,c+1 wrap is offset-neutral since 14*3584 == 50176).
// ---------------------------------------------------------------------------
__global__ __launch_bounds__(256) void k_sumsq(const float* __restrict__ x,
                                               float* __restrict__ sq) {
    const int b    = blockIdx.x >> 4;
    const int pi   = blockIdx.x & 15;
    const int tid  = threadIdx.x;
    const int wave = tid >> 5;
    const int lane = tid & 31;
    const int pj   = lane & 15;
    const int half = lane >> 4;

    __shared__ float s_sq[16];
    if (tid < 16) s_sq[tid] = 0.0f;
    __syncthreads();

    // lane-fixed column offset folded into the base pointer
    const float* px = x + (size_t)b * (CH * HW * HW) + pi * HW + pj;

    // the two B-matrix slots of this lane cover units u0 = it*32 + slot,
    // u1 = u0 + 2;  slots {wave*4+half} U {wave*4+half+2} tile [0,32).
    const int u0 = wave * 4 + half;   // [0, 29]
    const int u1 = u0 + 2;            // [2, 31]
    int gi0 = u0 / 14, gj0 = u0 - gi0 * 14;
    int gi1 = u1 / 14, gj1 = u1 - gi1 * 14;
    int off0 = gi0 * (16 * HW) + gj0 * 16;
    int off1 = gi1 * (16 * HW) + gj1 * 16;

    v2f a; a.x = 1.0f; a.y = 1.0f;    // all-ones A matrix (16x4)
    v8f acc = {};

    const int iters = (CH * GRID_G * GRID_G) / 32;  // 588
    for (int it = 0; it < iters; ++it) {
        const float e0 = px[off0];
        const float e1 = px[off1];
        gj0 += 4; off0 += 7232; if (gj0 >= 14) { gj0 -= 14; off0 += 3360; }
        gj1 += 4; off1 += 7232; if (gj1 >= 14) { gj1 -= 14; off1 += 3360; }
        v2f bm; bm.x = e0 * e0; bm.y = e1 * e1;
        acc = __builtin_amdgcn_wmma_f32_16x16x4_f32(false, a, false, bm,
                                                    (short)0, acc, false, false);
    }

    // D row 0 lives in acc[0], lanes 0..15 (N = pj).  Combine 8 waves via LDS.
    if (lane < 16) atomicAdd(&s_sq[pj], acc[0]);
    __syncthreads();
    if (tid < 16) sq[(blockIdx.x << 4) + tid] = s_sq[tid];
}

// ---------------------------------------------------------------------------
// Pass 2: per-batch argmax over 256 offsets, first-index tie-break.
// ---------------------------------------------------------------------------
__global__ __launch_bounds__(256) void k_argmax(const float* __restrict__ sq,
                                                int* __restrict__ shifts) {
    const int b = blockIdx.x;
    const int t = threadIdx.x;
    __shared__ float sv[256];
    __shared__ int   sidx[256];
    sv[t]   = sq[b * 256 + t];
    sidx[t] = t;
    __syncthreads();
    for (int off = 128; off > 0; off >>= 1) {
        if (t < off) {
            const float v2 = sv[t + off];
            const int   i2 = sidx[t + off];
            if (v2 > sv[t] || (v2 == sv[t] && i2 < sidx[t])) { sv[t] = v2; sidx[t] = i2; }
        }
        __syncthreads();
    }
    if (t == 0) {
        shifts[b * 2 + 0] = sidx[0] >> 4;  // si
        shifts[b * 2 + 1] = sidx[0] & 15;  // sj
    }
}

// ---------------------------------------------------------------------------
// Pass 3: circular roll.  out[b,c,h,w] = x[b,c,(h+si)%224,(w+sj)%224].
// Pure 32-bit index math (flat float4 index < 2^31); non-temporal loads and
// stores since the 1.85 GB stream has no reuse window inside the 192 MB L2.
// ---------------------------------------------------------------------------
__global__ __launch_bounds__(256) void k_roll(const float* __restrict__ x,
                                              const int* __restrict__ shifts,
                                              float* __restrict__ out) {
    const int i4 = blockIdx.x * 256 + threadIdx.x;  // float4 index, < 38,535,168
    const int w4 = i4 % (HW / 4);
    const int r  = i4 / (HW / 4);     // flat [b][c][h], < 688,128
    const int h  = r % HW;
    const int bc = r / HW;            // b*CH + c, < 3072
    const int b  = bc / CH;

    const int si = shifts[b * 2 + 0];
    const int sj = shifts[b * 2 + 1];
    int hr = h + si; if (hr >= HW) hr -= HW;

    const float* src = x + (size_t)bc * (HW * HW) + hr * HW;
    const int w = w4 * 4;
    int w0 = w + 0 + sj; if (w0 >= HW) w0 -= HW;
    int w1 = w + 1 + sj; if (w1 >= HW) w1 -= HW;
    int w2 = w + 2 + sj; if (w2 >= HW) w2 -= HW;
    int w3 = w + 3 + sj; if (w3 >= HW) w3 -= HW;

    v4f v;
    v.x = __builtin_nontemporal_load(src + w0);
    v.y = __builtin_nontemporal_load(src + w1);
    v.z = __builtin_nontemporal_load(src + w2);
    v.w = __builtin_nontemporal_load(src + w3);
    __builtin_nontemporal_store(v, reinterpret_cast<v4f*>(out) + i4);
}

// ---------------------------------------------------------------------------
extern "C" void kernel_launch(void* const* d_in, const int* in_sizes, int n_in,
                              void* d_out, int out_size, void* d_ws, size_t ws_size,
                              hipStream_t stream) {
    const float* x = (const float*)d_in[0];
    float* out = (float*)d_out;

    float* sq     = (float*)d_ws;                                      // 32 KB
    int*   shifts = (int*)((char*)d_ws + BATCH * 256 * sizeof(float)); // 32 * int2

    k_sumsq<<<BATCH * PATCH_P, 256, 0, stream>>>(x, sq);
    k_argmax<<<BATCH, 256, 0, stream>>>(sq, shifts);

    const int n4 = BATCH * CH * HW * (HW / 4);                 // 38,535,168
    k_roll<<<n4 / 256, 256, 0, stream>>>(x, shifts, out);      // 150,528 blocks
}